// Entmax15_88476326298033
// MI455X (gfx1250) — compile-verified
//
#include <hip/hip_runtime.h>
#include <cstdint>

// CDNA5 sparsemax: one workgroup (256 threads = 8 wave32) per 4096-float row.
// HBM traffic is the 11us floor (256 MiB @ 23.3 TB/s); everything else in LDS.
//  Stage 1: async global->LDS row load  (GLOBAL_LOAD_ASYNC_TO_LDS_B128, ASYNCcnt)
//  Stage 2: bitonic sort (descending) in LDS
//  Stage 3: cumsum via f32 WMMA triangular matmul P = X * U  (v_wmma_f32_16x16x4_f32)
//  Stage 4: support mask / masked-cumsum reduction, tau, write max(x - tau, 0)

typedef float v2f __attribute__((ext_vector_type(2)));
typedef float v8f __attribute__((ext_vector_type(8)));

#define N     4096
#define NBLK  256

__device__ __forceinline__ uint32_t lds_addr32(const void* p) {
  // generic pointers to LDS encode the LDS byte offset in the low 32 bits
  return (uint32_t)(uintptr_t)p;
}

__global__ __launch_bounds__(NBLK) void sparsemax_rows(const float* __restrict__ logits,
                                                       float* __restrict__ out,
                                                       int rows) {
  __shared__ float vals[N];      // original row (kept for final subtraction)
  __shared__ float srt[N];       // sorted descending
  __shared__ float totals[256];  // per-16-element-row sums -> inclusive scan
  __shared__ float redf[NBLK];
  __shared__ int   redi[NBLK];

  const unsigned tid = threadIdx.x;
  const int row = blockIdx.x;
  if (row >= rows) return;  // uniform per block

  const float* grow = logits + (size_t)row * N;

  // ---------------- Stage 1: async global -> LDS (16 KB row) ----------------
#pragma unroll
  for (int q = 0; q < 4; ++q) {
    const int f4 = (int)tid * 4 + q;            // float4 index within row
    const float* g = grow + f4 * 4;
    uint32_t loff = lds_addr32(&vals[f4 * 4]);
    asm volatile("global_load_async_to_lds_b128 %0, %1, off"
                 :: "v"(loff), "v"(g) : "memory");
  }
  asm volatile("s_wait_asynccnt 0x0" ::: "memory");
  __syncthreads();

  // copy vals -> srt (b128 LDS traffic)
  {
    const float4* v4 = (const float4*)vals;
    float4* s4 = (float4*)srt;
#pragma unroll
    for (int q = 0; q < 4; ++q) {
      const int f4 = (int)tid * 4 + q;
      s4[f4] = v4[f4];
    }
  }

  // ---------------- Stage 2: bitonic sort, descending -----------------------
  for (unsigned size = 2; size <= N; size <<= 1) {
    for (unsigned stride = size >> 1; stride > 0; stride >>= 1) {
      __syncthreads();
      for (unsigned cmp = tid; cmp < N / 2; cmp += NBLK) {
        const unsigned pos = 2u * cmp - (cmp & (stride - 1));
        const bool asc = (cmp & (size >> 1)) != 0;  // final dir = descending
        const float a = srt[pos];
        const float b = srt[pos + stride];
        const bool sw = asc ? (a > b) : (a < b);
        if (sw) { srt[pos] = b; srt[pos + stride] = a; }
      }
    }
  }
  __syncthreads();

  // ---------------- Stage 3: cumsum via WMMA triangular matmul ---------------
  // View sorted row as 256 rows x 16 cols. Each wave scans 2 tiles of 16 rows:
  //   P = X * U, U[k][j] = (j >= k), via 4 chained 16x16x4 f32 WMMAs.
  const int wave = tid >> 5;
  const int lane = tid & 31;
  const int half = lane >> 4;   // K-half selector for A/B operand layout
  const int nn   = lane & 15;   // N (column) index in B/C/D layout

  v8f pacc[2];
#pragma unroll
  for (int tt = 0; tt < 2; ++tt) {
    const int tile = wave * 2 + tt;       // 16 tiles of 16 rows
    v8f acc = {};
#pragma unroll
    for (int kb = 0; kb < 4; ++kb) {
      const int kcol = 4 * kb + 2 * half;
      // A (16x4 f32): lanes 0-15 -> M=lane, VGPR0/1 = K pair; lanes 16-31 -> K+2
      const float* ap = &srt[(tile * 16 + nn) * 16 + kcol];
      v2f a;  a.x = ap[0];  a.y = ap[1];
      // B (4x16 f32): constant rows of U; assumed symmetric layout to A
      v2f bm;
      bm.x = (nn >= kcol + 0) ? 1.0f : 0.0f;
      bm.y = (nn >= kcol + 1) ? 1.0f : 0.0f;
      acc = __builtin_amdgcn_wmma_f32_16x16x4_f32(
          /*neg_a=*/false, a, /*neg_b=*/false, bm,
          /*c_mod=*/(short)0, acc, /*reuse_a=*/false, /*reuse_b=*/false);
    }
    pacc[tt] = acc;
  }

  // row totals = within-row prefix at column 15 (held by lanes 15 and 31)
  if (nn == 15) {
#pragma unroll
    for (int tt = 0; tt < 2; ++tt) {
      const int rbase = (wave * 2 + tt) * 16 + half * 8;
#pragma unroll
      for (int v = 0; v < 8; ++v) totals[rbase + v] = pacc[tt][v];
    }
  }
  __syncthreads();

  // inclusive Hillis-Steele scan of the 256 row totals
  for (int d = 1; d < 256; d <<= 1) {
    const float y = (tid >= (unsigned)d) ? totals[tid - d] : 0.0f;
    __syncthreads();
    totals[tid] += y;
    __syncthreads();
  }

  // ---------------- Stage 4: support mask, tau, output -----------------------
  float tpart = 0.0f;
  int   kpart = 0;
#pragma unroll
  for (int tt = 0; tt < 2; ++tt) {
    const int tile = wave * 2 + tt;
#pragma unroll
    for (int v = 0; v < 8; ++v) {
      const int r   = tile * 16 + half * 8 + v;         // row of 16
      const float o = (r > 0) ? totals[r - 1] : 0.0f;   // exclusive row offset
      const float cs = pacc[tt][v] + o;                 // full inclusive cumsum
      const int i   = r * 16 + nn;                      // 0-based sorted index
      const float z = srt[i];
      if (z * (float)(i + 1) > cs - 1.0f) { tpart += cs; ++kpart; }
    }
  }
  redf[tid] = tpart;
  redi[tid] = kpart;
  __syncthreads();
  for (int s = NBLK / 2; s > 0; s >>= 1) {
    if ((int)tid < s) { redf[tid] += redf[tid + s]; redi[tid] += redi[tid + s]; }
    __syncthreads();
  }
  const float tau = (redf[0] - 1.0f) / fmaxf((float)redi[0], 1.0f);

  float4* orow = (float4*)(out + (size_t)row * N);
  const float4* v4 = (const float4*)vals;
#pragma unroll
  for (int q = 0; q < 4; ++q) {
    const int f4 = (int)tid * 4 + q;
    const float4 v = v4[f4];
    float4 o;
    o.x = fmaxf(v.x - tau, 0.0f);
    o.y = fmaxf(v.y - tau, 0.0f);
    o.z = fmaxf(v.z - tau, 0.0f);
    o.w = fmaxf(v.w - tau, 0.0f);
    orow[f4] = o;
  }
}

extern "C" void kernel_launch(void* const* d_in, const int* in_sizes, int n_in,
                              void* d_out, int out_size, void* d_ws, size_t ws_size,
                              hipStream_t stream) {
  const float* logits = (const float*)d_in[0];
  float* out = (float*)d_out;
  const int rows = in_sizes[0] / N;  // 8192 for (4, 2048, 4096)
  sparsemax_rows<<<dim3(rows), dim3(NBLK), 0, stream>>>(logits, out, rows);
}